// GCN_9122510536959
// MI455X (gfx1250) — compile-verified
//
#include <hip/hip_runtime.h>

typedef __attribute__((ext_vector_type(2))) float v2f;
typedef __attribute__((ext_vector_type(8))) float v8f;

// ---------------- utility kernels ----------------

__global__ void zero_f32(float* __restrict__ p, long long n) {
  long long i = (long long)blockIdx.x * blockDim.x + threadIdx.x;
  long long stride = (long long)gridDim.x * blockDim.x;
  for (; i < n; i += stride) p[i] = 0.0f;
}

__global__ void degree_kernel(const int* __restrict__ dst, float* __restrict__ deg, int nE) {
  int i = blockIdx.x * blockDim.x + threadIdx.x;
  if (i < nE) atomicAdd(&deg[dst[i]], 1.0f);
}

__global__ void dis_kernel(float* __restrict__ d, int n) {
  int i = blockIdx.x * blockDim.x + threadIdx.x;
  if (i < n) d[i] = rsqrtf(d[i] + 1.0f);
}

// ---------------- WMMA GEMM: H[nRows x NOUT] = X[nRows x 64] @ W[64 x NOUT] ----------------
// One wave per 16-row tile; NOUT/16 C-tiles per wave; K=64 in steps of 4 using
// V_WMMA_F32_16X16X4_F32 (full fp32, no precision loss vs reference).

template <int NOUT>
__global__ __launch_bounds__(256) void gemm_wmma_kernel(const float* __restrict__ X,
                                                        const float* __restrict__ W,
                                                        float* __restrict__ H, int nRows) {
  __shared__ float sW[64 * NOUT];
  for (int i = threadIdx.x; i < 64 * NOUT; i += blockDim.x) sW[i] = W[i];
  __syncthreads();

  const int wave = threadIdx.x >> 5;
  const int lane = threadIdx.x & 31;
  const int half = lane >> 4;   // lane-half selects K offset +2 (A/B) / M offset +8 (C)
  const int sub  = lane & 15;   // M index (A) / N index (B,C)
  const int row0 = (blockIdx.x * 8 + wave) * 16;
  if (row0 >= nRows) return;    // wave-uniform exit: EXEC stays all-ones for WMMA

  const float* xr = X + (long long)(row0 + sub) * 64;

  v8f acc[NOUT / 16];
#pragma unroll
  for (int t = 0; t < NOUT / 16; ++t) acc[t] = (v8f){0, 0, 0, 0, 0, 0, 0, 0};

#pragma unroll
  for (int k0 = 0; k0 < 64; k0 += 4) {
    const int ka = k0 + 2 * half;
    v2f a;
    a.x = xr[ka];
    a.y = xr[ka + 1];
#pragma unroll
    for (int t = 0; t < NOUT / 16; ++t) {
      v2f b;
      b.x = sW[ka * NOUT + t * 16 + sub];
      b.y = sW[(ka + 1) * NOUT + t * 16 + sub];
      // 8 args: (neg_a, A, neg_b, B, c_mod, C, reuse_a, reuse_b)
      acc[t] = __builtin_amdgcn_wmma_f32_16x16x4_f32(false, a, false, b, (short)0, acc[t],
                                                     false, false);
    }
  }

#pragma unroll
  for (int t = 0; t < NOUT / 16; ++t) {
#pragma unroll
    for (int r = 0; r < 8; ++r) {
      H[(long long)(row0 + r + 8 * half) * NOUT + t * 16 + sub] = acc[t][r];
    }
  }
}

// ---------------- edge scatter: agg[dst] += h[src] * dis[src]*dis[dst] ----------------
// 16 (or 8) feature-threads per edge -> coalesced float4 gather of h[src], 4 f32 L2 atomics.

template <int D>
__global__ __launch_bounds__(256) void scatter_kernel(const float* __restrict__ H,
                                                      const int* __restrict__ src,
                                                      const int* __restrict__ dst,
                                                      const float* __restrict__ dis,
                                                      float* __restrict__ agg,
                                                      long long nWork) {
  long long t = (long long)blockIdx.x * blockDim.x + threadIdx.x;
  if (t >= nWork) return;
  const int G = D / 4;
  long long e = t / G;
  int c = (int)(t - e * G) * 4;
  int s = src[e];
  int d = dst[e];
  float norm = dis[s] * dis[d];
  const float4 hv = *(const float4*)(H + (long long)s * D + c);
  float* o = agg + (long long)d * D + c;
  atomicAdd(o + 0, hv.x * norm);
  atomicAdd(o + 1, hv.y * norm);
  atomicAdd(o + 2, hv.z * norm);
  atomicAdd(o + 3, hv.w * norm);
}

// ---------------- fused self-loop + bias (+ReLU): agg = f(agg + h*dis^2 + b) ----------------

template <bool RELU>
__global__ __launch_bounds__(256) void combine_kernel(float* __restrict__ agg,
                                                      const float* __restrict__ H,
                                                      const float* __restrict__ dis,
                                                      const float* __restrict__ bias,
                                                      long long n, int D) {
  long long t = (long long)blockIdx.x * blockDim.x + threadIdx.x;
  if (t >= n) return;
  long long node = t / D;
  int c = (int)(t - node * D);
  float s = dis[node];
  float v = agg[t] + H[t] * s * s + bias[c];
  agg[t] = RELU ? fmaxf(v, 0.0f) : v;
}

// ---------------- launch ----------------

extern "C" void kernel_launch(void* const* d_in, const int* in_sizes, int n_in,
                              void* d_out, int out_size, void* d_ws, size_t ws_size,
                              hipStream_t stream) {
  const float* x  = (const float*)d_in[0];
  const int*   ei = (const int*)d_in[1];
  const float* W1 = (const float*)d_in[2];
  const float* b1 = (const float*)d_in[3];
  const float* W2 = (const float*)d_in[4];
  const float* b2 = (const float*)d_in[5];
  const float* W3 = (const float*)d_in[6];
  const float* b3 = (const float*)d_in[7];
  float* out = (float*)d_out;

  const int nN = in_sizes[0] / 64;      // 100000
  const int nE = in_sizes[1] / 2;       // 1600000
  const int* src = ei;
  const int* dst = ei + nE;

  float* ws   = (float*)d_ws;
  float* dis  = ws;                                   // nN floats (deg -> dis in place)
  float* bufA = ws + 100352;                          // nN*64 floats (h)
  float* bufB = bufA + (long long)nN * 64;            // nN*64 floats (agg / layer out)

  const int tiles   = (nN + 15) / 16;
  const int gblocks = (tiles + 7) / 8;                // 8 waves per block
  const long long nd64 = (long long)nN * 64;
  const long long nd32 = (long long)nN * 32;
  const long long w64  = (long long)nE * 16;          // edge work items, D=64
  const long long w32  = (long long)nE * 8;           // edge work items, D=32

  // degree + symmetric norm
  zero_f32<<<1024, 256, 0, stream>>>(dis, nN);
  degree_kernel<<<(nE + 255) / 256, 256, 0, stream>>>(dst, dis, nE);
  dis_kernel<<<(nN + 255) / 256, 256, 0, stream>>>(dis, nN);

  // layer 1: h=x@W1 ; agg ; relu(agg + h*dis^2 + b1)
  gemm_wmma_kernel<64><<<gblocks, 256, 0, stream>>>(x, W1, bufA, nN);
  zero_f32<<<4096, 256, 0, stream>>>(bufB, nd64);
  scatter_kernel<64><<<(int)((w64 + 255) / 256), 256, 0, stream>>>(bufA, src, dst, dis, bufB, w64);
  combine_kernel<true><<<(int)((nd64 + 255) / 256), 256, 0, stream>>>(bufB, bufA, dis, b1, nd64, 64);

  // layer 2
  gemm_wmma_kernel<64><<<gblocks, 256, 0, stream>>>(bufB, W2, bufA, nN);
  zero_f32<<<4096, 256, 0, stream>>>(bufB, nd64);
  scatter_kernel<64><<<(int)((w64 + 255) / 256), 256, 0, stream>>>(bufA, src, dst, dis, bufB, w64);
  combine_kernel<true><<<(int)((nd64 + 255) / 256), 256, 0, stream>>>(bufB, bufA, dis, b2, nd64, 64);

  // layer 3 (no relu), scatter straight into d_out
  gemm_wmma_kernel<32><<<gblocks, 256, 0, stream>>>(bufB, W3, bufA, nN);
  zero_f32<<<4096, 256, 0, stream>>>(out, nd32);
  scatter_kernel<32><<<(int)((w32 + 255) / 256), 256, 0, stream>>>(bufA, src, dst, dis, out, w32);
  combine_kernel<false><<<(int)((nd32 + 255) / 256), 256, 0, stream>>>(out, bufA, dis, b3, nd32, 32);
}